// OnnxGridSample_64699387346959
// MI455X (gfx1250) — compile-verified
//
#include <hip/hip_runtime.h>
#include <cstdint>

#define BATCH 16
#define CCH   64
#define HH    256
#define WW    256
#define HW    (HH * WW)      // 65536 elements per (b,c) slice
#define GPTS  (HH * WW)      // Hg*Wg grid points per batch
#define TPB   256

// 8-byte row-pair, only 4-byte aligned in general
struct __attribute__((packed)) fpair { float a, b; };

typedef float v2f __attribute__((ext_vector_type(2)));

__global__ __launch_bounds__(TPB) void grid_sample_bilinear_kernel(
    const float* __restrict__ x, const float* __restrict__ grid,
    float* __restrict__ out)
{
    __shared__ __align__(8) float sgrid[TPB * 2];

    const int t = threadIdx.x;
    const int b = blockIdx.y;
    const int p = blockIdx.x * TPB + t;          // grid point index in batch b

    // ---- CDNA5 async DMA: stage this block's grid coords global -> LDS ----
    const unsigned lds_off = (unsigned)(uintptr_t)(&sgrid[0]) + (unsigned)t * 8u;
    {
        uint64_t gaddr = (uint64_t)(uintptr_t)(grid + ((size_t)b * GPTS + (size_t)p) * 2);
        asm volatile("global_load_async_to_lds_b64 %0, %1, off"
                     :: "v"(lds_off), "v"(gaddr) : "memory");
        asm volatile("s_wait_asynccnt 0" ::: "memory");
    }
    // read back own 8-byte slot via the LDS path (ds_load_b64)
    v2f g;
    asm volatile("ds_load_b64 %0, %1\n\t"
                 "s_wait_dscnt 0"
                 : "=v"(g) : "v"(lds_off) : "memory");
    const float gx = g.x;
    const float gy = g.y;

    // unnormalize (align_corners=false): ix = ((gx+1)*W - 1) * 0.5 = gx*128 + 127.5
    const float ix = fmaf(gx, 0.5f * (float)WW, 0.5f * (float)WW - 0.5f);
    const float iy = fmaf(gy, 0.5f * (float)HH, 0.5f * (float)HH - 0.5f);

    const float fx0 = floorf(ix), fy0 = floorf(iy);
    const float wx1 = ix - fx0, wx0 = 1.0f - wx1;
    const float wy1 = iy - fy0, wy0 = 1.0f - wy1;

    const int ix0 = (int)fx0, iy0 = (int)fy0;
    const int ix1 = ix0 + 1,  iy1 = iy0 + 1;

    // zeros padding: mask weights for OOB corners
    const float mx0 = ((unsigned)ix0 < (unsigned)WW) ? 1.0f : 0.0f;
    const float mx1 = ((unsigned)ix1 < (unsigned)WW) ? 1.0f : 0.0f;
    const float my0 = ((unsigned)iy0 < (unsigned)HH) ? 1.0f : 0.0f;
    const float my1 = ((unsigned)iy1 < (unsigned)HH) ? 1.0f : 0.0f;

    const float w00 = wy0 * wx0 * (my0 * mx0);
    const float w01 = wy0 * wx1 * (my0 * mx1);
    const float w10 = wy1 * wx0 * (my1 * mx0);
    const float w11 = wy1 * wx1 * (my1 * mx1);

    // clamped indices
    const int cx0 = min(max(ix0, 0), WW - 1);
    const int cx1 = min(max(ix1, 0), WW - 1);
    const int cy0 = min(max(iy0, 0), HH - 1);
    const int cy1 = min(max(iy1, 0), HH - 1);

    // Row-pair trick: load {xc[px], xc[px+1]} once per row.
    // After clamping, cx0/cx1 are always px or px+1, so fold the select into weights.
    const int  px = min(cx0, WW - 2);
    const bool s0 = (cx0 != px);   // corner0 value comes from .b
    const bool s1 = (cx1 != px);   // corner1 value comes from .b

    const float wa0 = (s0 ? 0.0f : w00) + (s1 ? 0.0f : w01);  // weight on pair0.a
    const float wb0 = (s0 ? w00 : 0.0f) + (s1 ? w01 : 0.0f);  // weight on pair0.b
    const float wa1 = (s0 ? 0.0f : w10) + (s1 ? 0.0f : w11);  // weight on pair1.a
    const float wb1 = (s0 ? w10 : 0.0f) + (s1 ? w11 : 0.0f);  // weight on pair1.b

    const int o0 = cy0 * WW + px;   // element offset of row-y0 pair within slice
    const int o1 = cy1 * WW + px;   // element offset of row-y1 pair within slice

    const float* __restrict__ xb = x   + (size_t)b * CCH * HW;
    float* __restrict__       ob = out + (size_t)b * CCH * HW + (size_t)p;

    #pragma unroll 8
    for (int c = 0; c < CCH; ++c) {
        const float* xc = xb + (size_t)c * HW;
        const fpair r0 = *reinterpret_cast<const fpair*>(xc + o0);
        const fpair r1 = *reinterpret_cast<const fpair*>(xc + o1);
        const float res = fmaf(wa0, r0.a,
                          fmaf(wb0, r0.b,
                          fmaf(wa1, r1.a, wb1 * r1.b)));
        // streamed 256 MB output: non-temporal, keep L2 for x slices
        __builtin_nontemporal_store(res, ob + (size_t)c * HW);
    }
}

extern "C" void kernel_launch(void* const* d_in, const int* in_sizes, int n_in,
                              void* d_out, int out_size, void* d_ws, size_t ws_size,
                              hipStream_t stream) {
    const float* x    = (const float*)d_in[0];   // (16, 64, 256, 256) f32
    const float* grid = (const float*)d_in[1];   // (16, 256, 256, 2)  f32
    float* out        = (float*)d_out;           // (16, 64, 256, 256) f32

    dim3 gridDim(GPTS / TPB, BATCH, 1);          // (256, 16)
    grid_sample_bilinear_kernel<<<gridDim, TPB, 0, stream>>>(x, grid, out);
}